// VolumeGenerator_27986006900997
// MI455X (gfx1250) — compile-verified
//
#include <hip/hip_runtime.h>

typedef __attribute__((ext_vector_type(2))) float v2f;
typedef __attribute__((ext_vector_type(8))) float v8f;

#define BB   2
#define VV   4
#define CINC 256
#define HH   96
#define WW   96
#define COUT 32
#define HWP  (HH * WW)          // 9216 pixels per (b,v)
#define NPTS (64 * 64 * 64)     // 262144 voxels per batch

// -----------------------------------------------------------------------------
// Kernel 1: 1x1 conv as WMMA f32 GEMM, output in channels-last layout
//   feats_cl[bv][p][o] = sum_c features[bv][c][p] * W[o][c] + bias[o]
// One wave: 16 pixels (M) x 32 out-channels (two 16x16 N tiles), K=256 in
// steps of 4 via v_wmma_f32_16x16x4_f32.
// -----------------------------------------------------------------------------
__global__ __launch_bounds__(128)
void vg_conv1x1_wmma(const float* __restrict__ F,    // [bv][CIN][HW]
                     const float* __restrict__ Wp,   // [COUT][CIN]
                     const float* __restrict__ bp,   // [COUT]
                     float* __restrict__ feats_cl) { // [bv][HW][COUT]
    // Stage W transposed into LDS: Wt[k][o]  (32 KB)
    __shared__ float Wt[CINC * COUT];
    for (int i = threadIdx.x; i < CINC * COUT; i += blockDim.x) {
        int o = i & (COUT - 1);
        int k = i >> 5;
        Wt[k * COUT + o] = Wp[o * CINC + k];
    }
    __syncthreads();

    const int lane = threadIdx.x & 31;
    const int wave = blockIdx.x * (blockDim.x >> 5) + (threadIdx.x >> 5);
    const int tilesPerBV = HWP / 16;              // 576
    const int bv   = wave / tilesPerBV;           // 0..7
    const int p0   = (wave % tilesPerBV) * 16;    // pixel tile base
    const int m    = lane & 15;                   // M index (pixel) / N index
    const int half = lane >> 4;                   // K-pair selector

    const float* Fbv = F + (size_t)bv * CINC * HWP;

    v8f c0 = {};  // out-channels 0..15
    v8f c1 = {};  // out-channels 16..31

    for (int k = 0; k < CINC; k += 4) {
        const int k0 = k + 2 * half;
        // A fragment (16x4 f32): lane m holds pixel p0+m; VGPR0=K0, VGPR1=K0+1
        v2f a;
        a.x = Fbv[(size_t)(k0    ) * HWP + p0 + m];
        a.y = Fbv[(size_t)(k0 + 1) * HWP + p0 + m];
        // B fragments (4x16 f32), from LDS W^T
        v2f b0, b1;
        b0.x = Wt[(k0    ) * COUT + m];
        b0.y = Wt[(k0 + 1) * COUT + m];
        b1.x = Wt[(k0    ) * COUT + 16 + m];
        b1.y = Wt[(k0 + 1) * COUT + 16 + m];
        c0 = __builtin_amdgcn_wmma_f32_16x16x4_f32(false, a, false, b0,
                                                   (short)0, c0, false, false);
        c1 = __builtin_amdgcn_wmma_f32_16x16x4_f32(false, a, false, b1,
                                                   (short)0, c1, false, false);
    }

    const float bias0 = bp[m];
    const float bias1 = bp[16 + m];
    float* obase = feats_cl + ((size_t)bv * HWP + p0) * COUT;
    // C/D layout: VGPR r -> M = r + 8*half, N = lane&15
#pragma unroll
    for (int r = 0; r < 8; ++r) {
        const int prow = r + 8 * half;
        obase[(size_t)prow * COUT + m]      = c0[r] + bias0;
        obase[(size_t)prow * COUT + 16 + m] = c1[r] + bias1;
    }
}

// -----------------------------------------------------------------------------
// Kernel 2: fused projection + bilinear sampling + softmax-over-views.
// One wave handles 32 consecutive voxels; lane == output channel (COUT==32).
// Each corner gather is a fully coalesced 128B request from the channels-last
// L2-resident intermediate. Output transposed through padded LDS so stores to
// out[b][c][n] are coalesced.
// -----------------------------------------------------------------------------
__global__ __launch_bounds__(256)
void vg_sample_softmax(const float* __restrict__ feats_cl, // [bv][HW][COUT]
                       const float* __restrict__ proj,     // [b][v][3][4]
                       const float* __restrict__ coords,   // [b][NPTS][3]
                       float* __restrict__ out) {          // [b][COUT][NPTS]
    __shared__ float sP[BB * VV * 12];
    __shared__ float tile[8][32 * 33];  // per-wave transpose tile (padded)

    for (int i = threadIdx.x; i < BB * VV * 12; i += blockDim.x)
        sP[i] = proj[i];
    __syncthreads();

    const int lane      = threadIdx.x & 31;
    const int waveInBlk = threadIdx.x >> 5;
    const int wave      = blockIdx.x * 8 + waveInBlk;
    const int wavesPerB = NPTS / 32;
    const int b  = wave / wavesPerB;
    const int n0 = (wave % wavesPerB) * 32;

    // each lane loads its own point's coords; broadcast later via shuffles
    const float* cp = coords + ((size_t)b * NPTS + n0) * 3;
    const float cx = cp[lane * 3 + 0];
    const float cy = cp[lane * 3 + 1];
    const float cz = cp[lane * 3 + 2];

    float* my = tile[waveInBlk];

    for (int i = 0; i < 32; ++i) {
        const float X = __shfl(cx, i, 32);
        const float Y = __shfl(cy, i, 32);
        const float Z = __shfl(cz, i, 32);

        float s[VV];
#pragma unroll
        for (int v = 0; v < VV; ++v) {
            const float* P = sP + (b * VV + v) * 12;
            const float u  = P[0] * X + P[1] * Y + P[2]  * Z + P[3];
            const float ve = P[4] * X + P[5] * Y + P[6]  * Z + P[7];
            const float wz = P[8] * X + P[9] * Y + P[10] * Z + P[11];
            const bool invalid = (wz <= 0.0f);
            const float wzs = (wz == 0.0f) ? 1.0f : wz;
            const float uu = u / wzs;
            const float vv2 = ve / wzs;
            // faithful to reference: gx = 2*(u/H - .5), gy = 2*(v/W - .5)
            const float gx = 2.0f * (uu / (float)HH - 0.5f);
            const float gy = 2.0f * (vv2 / (float)WW - 0.5f);
            const float px = (gx + 1.0f) * 0.5f * (float)(WW - 1);
            const float py = (gy + 1.0f) * 0.5f * (float)(HH - 1);

            const float fx0 = floorf(px);
            const float fy0 = floorf(py);
            const int x0 = (int)fx0;
            const int y0 = (int)fy0;
            const float wx1 = px - fx0, wx0 = 1.0f - wx1;
            const float wy1 = py - fy0, wy0 = 1.0f - wy1;

            const float* fb = feats_cl + (size_t)(b * VV + v) * HWP * COUT;
            float acc = 0.0f;
            const int   xs[2]  = {x0, x0 + 1};
            const int   ys[2]  = {y0, y0 + 1};
            const float wxs[2] = {wx0, wx1};
            const float wys[2] = {wy0, wy1};
#pragma unroll
            for (int jy = 0; jy < 2; ++jy) {
#pragma unroll
                for (int jx = 0; jx < 2; ++jx) {
                    const int xi = xs[jx], yi = ys[jy];
                    if (xi >= 0 && xi < WW && yi >= 0 && yi < HH) {
                        // coalesced 128B gather: lane = channel
                        const float val =
                            fb[((size_t)yi * WW + xi) * COUT + lane];
                        acc += val * (wxs[jx] * wys[jy]);
                    }
                }
            }
            s[v] = invalid ? 0.0f : acc;
        }

        // softmax over the 4 views (per lane / channel), weighted sum
        const float mx = fmaxf(fmaxf(s[0], s[1]), fmaxf(s[2], s[3]));
        const float e0 = __expf(s[0] - mx);
        const float e1 = __expf(s[1] - mx);
        const float e2 = __expf(s[2] - mx);
        const float e3 = __expf(s[3] - mx);
        const float denom = e0 + e1 + e2 + e3;
        const float vol =
            (s[0] * e0 + s[1] * e1 + s[2] * e2 + s[3] * e3) / denom;

        my[i * 33 + lane] = vol;  // point i, channel lane
    }

    // wave-private LDS region; DS ops are in-order within a wave.
    // Coalesced store: for each channel, 32 consecutive voxels.
#pragma unroll 4
    for (int c = 0; c < COUT; ++c) {
        out[((size_t)(b * COUT + c)) * NPTS + n0 + lane] = my[lane * 33 + c];
    }
}

// -----------------------------------------------------------------------------
extern "C" void kernel_launch(void* const* d_in, const int* in_sizes, int n_in,
                              void* d_out, int out_size, void* d_ws,
                              size_t ws_size, hipStream_t stream) {
    (void)in_sizes; (void)n_in; (void)out_size; (void)ws_size;
    const float* features = (const float*)d_in[0];  // [B,V,CIN,H,W]
    const float* proj     = (const float*)d_in[1];  // [B,V,3,4]
    const float* coords   = (const float*)d_in[2];  // [B,64,64,64,3]
    const float* Wp       = (const float*)d_in[3];  // [COUT,CIN]
    const float* bp       = (const float*)d_in[4];  // [COUT]
    float* out      = (float*)d_out;                // [B,COUT,64,64,64]
    float* feats_cl = (float*)d_ws;                 // needs 9.44 MB

    // GEMM: 8 bv * 576 pixel-tiles = 4608 waves; 4 waves/block -> 1152 blocks
    vg_conv1x1_wmma<<<1152, 128, 0, stream>>>(features, Wp, bp, feats_cl);

    // Sampling: 2*262144/32 = 16384 waves; 8 waves/block -> 2048 blocks
    vg_sample_softmax<<<2048, 256, 0, stream>>>(feats_cl, proj, coords, out);
}